// BasicAttentionBlock_59339268161977
// MI455X (gfx1250) — compile-verified
//
#include <hip/hip_runtime.h>

typedef __attribute__((ext_vector_type(2))) float v2f;
typedef __attribute__((ext_vector_type(8))) float v8f;

#define N_TOK   49
#define HEADS   16
#define DHEAD   32
#define C_ALL   512
#define NW      1024

// LDS strides (in floats), chosen for bank-conflict-free half-wave access
#define KT_STRIDE  72   // kt[32][72]   : K^T, kt[k][n]
#define SV_STRIDE  48   // sV[64][48]   : V,   sV[n][c], padded rows zeroed
#define SP_STRIDE  68   // sP[64][68]   : normalized probabilities P
#define ADD_STRIDE 52   // sAdd[49][52] : bias[rel_index] + mask

__global__ __launch_bounds__(128, 1) void swin_attn_wmma_f32(
    const float* __restrict__ q,
    const float* __restrict__ k,
    const float* __restrict__ v,
    const float* __restrict__ bias_table,
    const float* __restrict__ mask,
    const int*   __restrict__ rel_index,
    float*       __restrict__ out)
{
    __shared__ float kt[32 * KT_STRIDE];
    __shared__ float sV[64 * SV_STRIDE];
    __shared__ float sP[64 * SP_STRIDE];
    __shared__ float sAdd[N_TOK * ADD_STRIDE];

    const int blk  = blockIdx.x;
    const int b    = blk >> 4;          // window-batch index (0..4095)
    const int h    = blk & 15;          // head
    const int w    = b & (NW - 1);      // mask window = b % 1024

    const int tid   = threadIdx.x;
    const int wave  = tid >> 5;         // 0..3 : row tile (rows 16*wave..16*wave+15)
    const int lane  = tid & 31;
    const int lhalf = lane >> 4;        // 0/1
    const int l16   = lane & 15;

    const float* qb = q + ((size_t)b * N_TOK) * C_ALL + h * DHEAD;
    const float* kb = k + ((size_t)b * N_TOK) * C_ALL + h * DHEAD;
    const float* vb = v + ((size_t)b * N_TOK) * C_ALL + h * DHEAD;

    // ---- zero ONLY the pad regions a WMMA B-operand can touch:
    //      kt columns n = 49..63 (k = 0..31), sV rows n = 49..63 (c = 0..31).
    //      Disjoint from the fill region below, so no barrier needed in between.
    for (int i = tid; i < 2 * 15 * 32; i += 128) {
        int second = (i >= 15 * 32);
        int j = i - second * (15 * 32);   // 0..479
        int r = j >> 5;                   // 0..14 -> index 49+r
        int c = j & 31;                   // 0..31
        if (!second) kt[c * KT_STRIDE + (N_TOK + r)] = 0.0f;
        else         sV[(N_TOK + r) * SV_STRIDE + c] = 0.0f;
    }

    // ---- stage K^T and V (coalesced global reads) ----
    for (int i = tid; i < N_TOK * DHEAD; i += 128) {
        int n = i >> 5;                 // token 0..48
        int c = i & 31;                 // channel 0..31
        kt[c * KT_STRIDE + n] = kb[n * C_ALL + c];
        sV[n * SV_STRIDE + c] = vb[n * C_ALL + c];
    }
    // ---- combined bias + mask matrix ----
    for (int i = tid; i < N_TOK * N_TOK; i += 128) {
        int m = i / N_TOK;
        int n = i - m * N_TOK;
        int ridx = rel_index[i];
        sAdd[m * ADD_STRIDE + n] =
            bias_table[ridx * HEADS + h] + mask[(size_t)w * (N_TOK * N_TOK) + i];
    }
    __syncthreads();

    // ---- preload Q rows in A-operand layout: lane supplies row M = l16, K pair (2*lhalf, +1) ----
    int m_a = wave * 16 + l16;
    int m_clamped = (m_a < N_TOK) ? m_a : (N_TOK - 1);   // clamp in-bounds; padded rows discarded
    const float* qrow = qb + m_clamped * C_ALL + 2 * lhalf;
    v2f aQ[8];
#pragma unroll
    for (int kk = 0; kk < 8; ++kk)
        aQ[kk] = *(const v2f*)(qrow + 4 * kk);           // 8B-aligned

    // ---- GEMM1: S = Q @ K^T, 4 column tiles x 8 k-chunks of 16x16x4 f32 WMMA ----
    v8f S[4];
#pragma unroll
    for (int ct = 0; ct < 4; ++ct) {
        v8f acc = {};
        int nb = ct * 16 + l16;                          // key column this lane supplies
#pragma unroll
        for (int kk = 0; kk < 8; ++kk) {
            int krow = kk * 4 + 2 * lhalf;
            v2f bK;
            bK.x = kt[krow * KT_STRIDE + nb];
            bK.y = kt[(krow + 1) * KT_STRIDE + nb];
            acc = __builtin_amdgcn_wmma_f32_16x16x4_f32(
                false, aQ[kk], false, bK, (short)0, acc, false, false);
        }
        S[ct] = acc;
    }

    // ---- epilogue: scale + bias + mask, row softmax, write normalized P to LDS ----
    const float scale = 0.17677669529663687f;            // 32^-0.5
#pragma unroll
    for (int vv = 0; vv < 8; ++vv) {
        int m = wave * 16 + vv + 8 * lhalf;              // row in C/D layout
        float mx = -3.0e38f;
#pragma unroll
        for (int ct = 0; ct < 4; ++ct) {
            int n = ct * 16 + l16;
            float s;
            if (m < N_TOK && n < N_TOK)
                s = S[ct][vv] * scale + sAdd[m * ADD_STRIDE + n];
            else
                s = -1.0e9f;                             // padded keys -> zero prob
            S[ct][vv] = s;
            mx = fmaxf(mx, s);
        }
        // row max across the 16 lanes of this half-wave
#pragma unroll
        for (int off = 1; off <= 8; off <<= 1)
            mx = fmaxf(mx, __shfl_xor(mx, off, 32));
        float sum = 0.0f;
#pragma unroll
        for (int ct = 0; ct < 4; ++ct) {
            float p = __expf(S[ct][vv] - mx);
            S[ct][vv] = p;
            sum += p;
        }
#pragma unroll
        for (int off = 1; off <= 8; off <<= 1)
            sum += __shfl_xor(sum, off, 32);
        float inv = 1.0f / sum;
#pragma unroll
        for (int ct = 0; ct < 4; ++ct)
            sP[m * SP_STRIDE + ct * 16 + l16] = S[ct][vv] * inv;
    }
    // same-wave LDS ops are in-order: the A-operand reads below see these writes

    // ---- GEMM2: O = P @ V, 16 k-chunks x 2 output tiles of 16x16x4 f32 WMMA ----
    v8f O0 = {}, O1 = {};
    int mrow = wave * 16 + l16;                          // A row this lane supplies
#pragma unroll
    for (int kk = 0; kk < 16; ++kk) {
        int kb4 = kk * 4 + 2 * lhalf;
        v2f aP = *(const v2f*)(&sP[mrow * SP_STRIDE + kb4]);   // 8B-aligned
        v2f bV0, bV1;
        bV0.x = sV[kb4 * SV_STRIDE + l16];
        bV0.y = sV[(kb4 + 1) * SV_STRIDE + l16];
        bV1.x = sV[kb4 * SV_STRIDE + 16 + l16];
        bV1.y = sV[(kb4 + 1) * SV_STRIDE + 16 + l16];
        O0 = __builtin_amdgcn_wmma_f32_16x16x4_f32(
            false, aP, false, bV0, (short)0, O0, false, false);
        O1 = __builtin_amdgcn_wmma_f32_16x16x4_f32(
            false, aP, false, bV1, (short)0, O1, false, false);
    }

    // ---- store O (coalesced 64B per half-wave) ----
    float* ob = out + ((size_t)b * N_TOK) * C_ALL + h * DHEAD;
#pragma unroll
    for (int vv = 0; vv < 8; ++vv) {
        int m = wave * 16 + vv + 8 * lhalf;
        if (m < N_TOK) {
            ob[m * C_ALL + l16]      = O0[vv];
            ob[m * C_ALL + 16 + l16] = O1[vv];
        }
    }
}

extern "C" void kernel_launch(void* const* d_in, const int* in_sizes, int n_in,
                              void* d_out, int out_size, void* d_ws, size_t ws_size,
                              hipStream_t stream) {
    const float* q          = (const float*)d_in[0];
    const float* k          = (const float*)d_in[1];
    const float* v          = (const float*)d_in[2];
    const float* bias_table = (const float*)d_in[3];
    const float* mask       = (const float*)d_in[4];
    const int*   rel_index  = (const int*)d_in[5];
    float* out = (float*)d_out;

    const int B_ = 4096;                 // NW * B = 1024 * 4
    dim3 grid(B_ * HEADS);               // one block per (window, head)
    swin_attn_wmma_f32<<<grid, 128, 0, stream>>>(q, k, v, bias_table, mask, rel_index, out);
}